// NCC_Loss_31525059953186
// MI455X (gfx1250) — compile-verified
//
#include <hip/hip_runtime.h>
#include <math.h>

typedef __attribute__((ext_vector_type(2))) float v2f;
typedef __attribute__((ext_vector_type(8))) float v8f;

#define KWIN   9
#define PAD    4
#define TILE   16
#define HALO   24          // TILE + 2*PAD
#define HVOL   (HALO*HALO*HALO)  // 13824
#define DIMS   128
#define NTILES 2048        // 4 batches * 8^3 tiles
#define NVOX   4194304.0f  // 2 * 128^3

__device__ __forceinline__ float bandv(int k, int m) {
    // Band[m][k] = 1 if output position m sums halo positions m..m+8
    return (k >= m && k <= m + (KWIN - 1)) ? 1.0f : 0.0f;
}

// field value; constant-folds after full unroll of the field loop
__device__ __forceinline__ float fval(int f, float i, float j) {
    switch (f) {
        case 0:  return i;
        case 1:  return j;
        case 2:  return i * i;
        case 3:  return j * j;
        default: return i * j;
    }
}

__global__ __launch_bounds__(256) void ncc_tile_kernel(
        const float* __restrict__ I, const float* __restrict__ J,
        float* __restrict__ partial)
{
    extern __shared__ float smem[];
    float* Ih = smem;                    // [24][24][24] I halo (zero-padded)
    float* Jh = Ih + HVOL;               // [24][24][24] J halo
    float* T1 = Jh + HVOL;               // [24][24][16]  (d, h, wo)
    float* T2 = T1 + HALO * HALO * TILE; // [24][16][16]  (d, ho, wo)
    float* F  = T2 + HALO * TILE * TILE; // [5][16][16][16] filtered fields
    __shared__ float wsum[8];

    const int tid   = threadIdx.x;
    const int lane  = tid & 31;
    const int wave  = tid >> 5;     // 0..7
    const int n16   = lane & 15;    // N column (B/C/D) or M row (A)
    const int khalf = lane >> 4;    // which K-half of each K=4 group
    const int mbase = khalf * 8;    // C/D row offset for this lane half

    // tile coordinates
    const int t  = blockIdx.x;
    const int b  = t >> 9;          // / 512
    const int r  = t & 511;
    const int td = r >> 6;
    const int rr = r & 63;
    const int th = rr >> 3;
    const int tw = rr & 7;
    const int d0 = td * TILE - PAD;
    const int h0 = th * TILE - PAD;
    const int w0 = tw * TILE - PAD;
    const long base = (long)b * DIMS * DIMS * DIMS;

    // ---------- Stage 0: cooperative zero-padded halo load (once) ----------
    for (int e = tid; e < HVOL; e += 256) {
        const int w  = e % HALO;
        const int e2 = e / HALO;
        const int h  = e2 % HALO;
        const int d  = e2 / HALO;
        const int gd = d0 + d, gh = h0 + h, gw = w0 + w;
        float vi = 0.f, vj = 0.f;
        if (((unsigned)gd < DIMS) & ((unsigned)gh < DIMS) & ((unsigned)gw < DIMS)) {
            const long idx = base + ((long)gd * DIMS + gh) * DIMS + gw;
            vi = I[idx];
            vj = J[idx];
        }
        Ih[e] = vi;
        Jh[e] = vj;
    }
    __syncthreads();

    #pragma unroll
    for (int f = 0; f < 5; ++f) {
        // ---------- Stage W: LDS halo -> T1, filter along W ----------
        // 48 jobs: d in [0,24) x h-chunk {0,1}; 6 per wave (uniform)
        for (int job = wave; job < HALO * 2; job += 8) {
            const int d  = job >> 1;
            const int hc = job & 1;
            const int h  = hc * 8 + n16;          // this lane's halo-h column
            const int rowoff = (d * HALO + h) * HALO;
            v8f c = {0.f, 0.f, 0.f, 0.f, 0.f, 0.f, 0.f, 0.f};
            #pragma unroll
            for (int ks = 0; ks < 6; ++ks) {
                const int kg = ks * 4 + 2 * khalf; // halo-w index of first elem
                v2f a, bb;
                a.x  = bandv(kg, n16);  a.y = bandv(kg + 1, n16);
                bb.x = fval(f, Ih[rowoff + kg    ], Jh[rowoff + kg    ]);
                bb.y = fval(f, Ih[rowoff + kg + 1], Jh[rowoff + kg + 1]);
                c = __builtin_amdgcn_wmma_f32_16x16x4_f32(
                        false, a, false, bb, (short)0, c, false, false);
            }
            // D[m][n]: m = wo, n = h-column; chunk1 stores only h >= 16
            if (hc == 0 || n16 >= 8) {
                #pragma unroll
                for (int i = 0; i < 8; ++i)
                    T1[(d * HALO + h) * TILE + (i + mbase)] = c[i];
            }
        }
        __syncthreads();

        // ---------- Stage H: T1 -> T2, filter along H ----------
        for (int d = wave; d < HALO; d += 8) {
            v8f c = {0.f, 0.f, 0.f, 0.f, 0.f, 0.f, 0.f, 0.f};
            #pragma unroll
            for (int ks = 0; ks < 6; ++ks) {
                const int kg = ks * 4 + 2 * khalf;
                v2f a, bb;
                a.x  = bandv(kg, n16);  a.y = bandv(kg + 1, n16);
                bb.x = T1[(d * HALO + kg    ) * TILE + n16];
                bb.y = T1[(d * HALO + kg + 1) * TILE + n16];
                c = __builtin_amdgcn_wmma_f32_16x16x4_f32(
                        false, a, false, bb, (short)0, c, false, false);
            }
            #pragma unroll
            for (int i = 0; i < 8; ++i)
                T2[(d * TILE + (i + mbase)) * TILE + n16] = c[i];
        }
        __syncthreads();

        // ---------- Stage D: T2 -> F[f], filter along D ----------
        for (int ho = wave; ho < TILE; ho += 8) {
            v8f c = {0.f, 0.f, 0.f, 0.f, 0.f, 0.f, 0.f, 0.f};
            #pragma unroll
            for (int ks = 0; ks < 6; ++ks) {
                const int kg = ks * 4 + 2 * khalf;
                v2f a, bb;
                a.x  = bandv(kg, n16);  a.y = bandv(kg + 1, n16);
                bb.x = T2[((kg    ) * TILE + ho) * TILE + n16];
                bb.y = T2[((kg + 1) * TILE + ho) * TILE + n16];
                c = __builtin_amdgcn_wmma_f32_16x16x4_f32(
                        false, a, false, bb, (short)0, c, false, false);
            }
            #pragma unroll
            for (int i = 0; i < 8; ++i)
                F[f * 4096 + ((i + mbase) * TILE + ho) * TILE + n16] = c[i];
        }
        __syncthreads();
    }

    // ---------- per-voxel cc + block reduction ----------
    float psum = 0.f;
    const float win = 729.0f;            // 9^3
    for (int v = tid; v < 4096; v += 256) {
        const float s0 = F[v];
        const float s1 = F[4096  + v];
        const float s2 = F[8192  + v];
        const float s3 = F[12288 + v];
        const float s4 = F[16384 + v];
        const float uI = s0 * (1.0f / win);
        const float uJ = s1 * (1.0f / win);
        const float cross = s4 - uJ * s0 - uI * s1 + uI * uJ * win;
        const float iv    = s2 - 2.f * uI * s0 + uI * uI * win;
        const float jv    = s3 - 2.f * uJ * s1 + uJ * uJ * win;
        psum += cross * cross / (iv * jv + 1e-5f);
    }
    #pragma unroll
    for (int off = 16; off > 0; off >>= 1)
        psum += __shfl_down(psum, off, 32);
    if (lane == 0) wsum[wave] = psum;
    __syncthreads();
    if (tid == 0) {
        float s = 0.f;
        #pragma unroll
        for (int i = 0; i < 8; ++i) s += wsum[i];
        partial[blockIdx.x] = s;   // deterministic: overwritten every call
    }
}

__global__ void ncc_finalize(const float* __restrict__ partial,
                             float* __restrict__ out)
{
    if (blockIdx.x == 0 && threadIdx.x == 0) {
        float s = 0.f;
        for (int i = 0; i < NTILES; ++i) s += partial[i];
        out[0] = -sqrtf(s / NVOX);
    }
}

extern "C" void kernel_launch(void* const* d_in, const int* in_sizes, int n_in,
                              void* d_out, int out_size, void* d_ws, size_t ws_size,
                              hipStream_t stream) {
    const float* I = (const float*)d_in[0];   // prediction
    const float* J = (const float*)d_in[1];   // target
    float* out     = (float*)d_out;
    float* partial = (float*)d_ws;            // NTILES floats (8 KB)

    const size_t shmem =
        (size_t)(2 * HVOL + HALO * HALO * TILE + HALO * TILE * TILE
                 + 5 * TILE * TILE * TILE) * sizeof(float);   // 253,952 B < 320 KB/WGP

    ncc_tile_kernel<<<NTILES, 256, shmem, stream>>>(I, J, partial);
    ncc_finalize<<<1, 1, 0, stream>>>(partial, out);
}